// EqualityConstrainedQuadratic_16441134809753
// MI455X (gfx1250) — compile-verified
//
#include <hip/hip_runtime.h>
#include <hip/hip_bf16.h>

typedef float v2f __attribute__((ext_vector_type(2)));
typedef float v8f __attribute__((ext_vector_type(8)));

#define N_DIM 512
#define E_DIM 64
#define KDIM  576          // N_DIM + E_DIM
#define LDK   576
#define NB    64           // panel/block size
#define NBLK  9            // KDIM / NB
#define BATCH 128

static __device__ __forceinline__ v8f wmma4(v2f a, v2f b, v8f c) {
  // V_WMMA_F32_16X16X4_F32 : D = A(16x4) * B(4x16) + C(16x16), fp32 throughout
  return __builtin_amdgcn_wmma_f32_16x16x4_f32(false, a, false, b, (short)0, c,
                                               false, false);
}

// 16x16 tile GEMM over k=64: all 16 fragment pairs are prefetched into
// registers (one wait for the whole batch), then the 16-deep WMMA chain runs
// without per-step memory stalls. A is row-major (lda), B row-major (ldb);
// pointers are tile origins. m16 = lane&15, khi = (lane>>4)*2 per the
// CDNA5 fp32 A/B fragment layouts.
static __device__ __forceinline__ v8f tile_gemm(const float* __restrict__ A, int lda,
                                                const float* __restrict__ B, int ldb,
                                                int m16, int khi) {
  v2f aF[16], bF[16];
#pragma unroll
  for (int q = 0; q < 16; ++q) {
    const int k = q * 4 + khi;
    aF[q].x = A[m16 * lda + k];
    aF[q].y = A[m16 * lda + k + 1];
    bF[q].x = B[k * ldb + m16];
    bF[q].y = B[(k + 1) * ldb + m16];
  }
  v8f acc = {};
#pragma unroll
  for (int q = 0; q < 16; ++q) acc = wmma4(aF[q], bF[q], acc);
  return acc;
}

// ---------------------------------------------------------------------------
// Kernel 1: assemble the (batch-invariant) KKT matrix
//   K = [[ M/2 + I , A0^T ],
//        [ A0      , 0    ]]
// ---------------------------------------------------------------------------
__global__ void build_kkt_kernel(const float* __restrict__ M,
                                 const float* __restrict__ A0,
                                 float* __restrict__ K) {
  int idx = blockIdx.x * blockDim.x + threadIdx.x;
  if (idx >= KDIM * KDIM) return;
  int i = idx / KDIM, j = idx - i * KDIM;
  float v;
  if (i < N_DIM && j < N_DIM) {
    v = 0.5f * M[i * N_DIM + j] + ((i == j) ? 1.0f : 0.0f);
  } else if (i < N_DIM) {                 // top-right: A0^T
    v = A0[(j - N_DIM) * N_DIM + i];
  } else if (j < N_DIM) {                 // bottom-left: A0
    v = A0[(i - N_DIM) * N_DIM + j];
  } else {
    v = 0.0f;
  }
  K[i * LDK + j] = v;
}

// ---------------------------------------------------------------------------
// Kernel 2: in-place blocked LU (no pivoting) of K, single workgroup.
//   Diagonal block factored in LDS; triangular inverses built in LDS so the
//   panel TRSMs and the rank-64 Schur trailing update all run as WMMA GEMMs.
// ---------------------------------------------------------------------------
__global__ __launch_bounds__(1024) void lu_factor_kernel(float* __restrict__ K) {
  __shared__ float sD [NB][NB + 1];
  __shared__ float sLi[NB][NB + 1];
  __shared__ float sUi[NB][NB + 1];

  const int tid  = threadIdx.x;
  const int lane = tid & 31;
  const int wave = tid >> 5;            // 32 waves
  const int m16  = lane & 15;
  const int khi  = (lane >> 4) * 2;     // A/B fragment k-pair select
  const int hi   = lane >> 4;

  for (int kb = 0; kb < NBLK; ++kb) {
    const int base = kb * NB;

    // ---- stage diagonal block in LDS ----
    for (int idx = tid; idx < NB * NB; idx += 1024) {
      int i = idx >> 6, j = idx & 63;
      sD[i][j] = K[(base + i) * LDK + (base + j)];
    }
    __syncthreads();

    // ---- unblocked LU of the 64x64 diagonal block ----
    for (int p = 0; p < NB - 1; ++p) {
      if (tid > p && tid < NB) sD[tid][p] /= sD[p][p];
      __syncthreads();
      for (int idx = tid; idx < NB * NB; idx += 1024) {
        int i = idx >> 6, j = idx & 63;
        if (i > p && j > p) sD[i][j] -= sD[i][p] * sD[p][j];
      }
      __syncthreads();
    }

    // ---- write back packed L\U diagonal block ----
    for (int idx = tid; idx < NB * NB; idx += 1024) {
      int i = idx >> 6, j = idx & 63;
      K[(base + i) * LDK + (base + j)] = sD[i][j];
    }

    // ---- triangular inverses, one column per thread ----
    if (tid < NB) {                      // Linv (unit lower)
      const int cc = tid;
      for (int i = 0; i < NB; ++i) {
        if (i < cc) { sLi[i][cc] = 0.0f; continue; }
        float s = (i == cc) ? 1.0f : 0.0f;
        for (int j = cc; j < i; ++j) s -= sD[i][j] * sLi[j][cc];
        sLi[i][cc] = s;
      }
    } else if (tid < 2 * NB) {           // Uinv
      const int cc = tid - NB;
      for (int i = NB - 1; i >= 0; --i) {
        if (i > cc) { sUi[i][cc] = 0.0f; continue; }
        float s = (i == cc) ? 1.0f : 0.0f;
        for (int j = i + 1; j <= cc; ++j) s -= sD[i][j] * sUi[j][cc];
        sUi[i][cc] = s / sD[i][i];
      }
    }
    __syncthreads();

    const int nt = NBLK - 1 - kb;        // trailing block count

    // ---- panel TRSMs as WMMA GEMMs (two-pass to avoid in-place hazard) ----
    // row panel: U(kb,jb) = Linv * K(kb,jb) ; col panel: L(ib,kb) = K(ib,kb) * Uinv
    {
      v8f accs[8];
      const int tot = nt * 16 * 2;
      int cnt = 0;
      for (int t = wave; t < tot; t += 32) {
        const bool rowP = (t < nt * 16);
        const int  tt   = t & 15;
        const int  blk  = (rowP ? t : t - nt * 16) >> 4;
        const int  tr   = (tt >> 2) * 16;
        const int  tc   = (tt & 3) * 16;
        if (rowP) {
          const int jb = (kb + 1 + blk) * NB;
          accs[cnt++] = tile_gemm(&sLi[tr][0], NB + 1,
                                  K + base * LDK + jb + tc, LDK, m16, khi);
        } else {
          const int ib = (kb + 1 + blk) * NB;
          accs[cnt++] = tile_gemm(K + (ib + tr) * LDK + base, LDK,
                                  &sUi[0][tc], NB + 1, m16, khi);
        }
      }
      __syncthreads();                   // all panel reads done before overwrite
      cnt = 0;
      for (int t = wave; t < tot; t += 32) {
        const bool rowP = (t < nt * 16);
        const int  tt   = t & 15;
        const int  blk  = (rowP ? t : t - nt * 16) >> 4;
        const int  tr   = (tt >> 2) * 16;
        const int  tc   = (tt & 3) * 16;
        v8f acc = accs[cnt++];
        if (rowP) {
          const int jb = (kb + 1 + blk) * NB;
          #pragma unroll
          for (int v = 0; v < 8; ++v)
            K[(base + tr + hi * 8 + v) * LDK + jb + tc + m16] = acc[v];
        } else {
          const int ib = (kb + 1 + blk) * NB;
          #pragma unroll
          for (int v = 0; v < 8; ++v)
            K[(ib + tr + hi * 8 + v) * LDK + base + tc + m16] = acc[v];
        }
      }
    }
    __syncthreads();

    // ---- trailing Schur update: C(ib,jb) -= L(ib,kb) * U(kb,jb) ----
    const int totC = nt * nt * 16;
    for (int t = wave; t < totC; t += 32) {
      const int tt  = t & 15;
      const int blk = t >> 4;
      const int bi  = blk / nt;
      const int bj  = blk - bi * nt;
      const int ib  = (kb + 1 + bi) * NB;
      const int jb  = (kb + 1 + bj) * NB;
      const int tr  = (tt >> 2) * 16;
      const int tc  = (tt & 3) * 16;
      v8f acc = tile_gemm(K + (ib + tr) * LDK + base, LDK,
                          K + base * LDK + jb + tc, LDK, m16, khi);
      #pragma unroll
      for (int v = 0; v < 8; ++v)
        K[(ib + tr + hi * 8 + v) * LDK + jb + tc + m16] -= acc[v];
    }
    __syncthreads();
  }
}

// ---------------------------------------------------------------------------
// Kernel 3: 128-RHS blocked forward/backward substitution.
//   Each workgroup owns 32 RHS columns fully resident in LDS; off-diagonal
//   block updates are WMMA GEMMs (A from L2-resident K, B from LDS).
// ---------------------------------------------------------------------------
__global__ __launch_bounds__(256) void kkt_solve_kernel(const float* __restrict__ K,
                                                        const float* __restrict__ x,
                                                        const float* __restrict__ parms,
                                                        const float* __restrict__ cvec,
                                                        float* __restrict__ out) {
  __shared__ float sX[KDIM][33];         // 576 x 32 RHS chunk (+pad)
  __shared__ float sDg[NB][NB + 1];      // staged diagonal block
  const int tid  = threadIdx.x;
  const int lane = tid & 31;
  const int wave = tid >> 5;             // 8 waves
  const int m16  = lane & 15;
  const int khi  = (lane >> 4) * 2;
  const int hi   = lane >> 4;
  const int col0 = blockIdx.x * 32;

  // RHS: top = x - parms (gamma = 1), bottom = -c
  for (int idx = tid; idx < KDIM * 32; idx += 256) {
    const int i  = idx >> 5;
    const int cc = idx & 31;
    const int b  = col0 + cc;
    sX[i][cc] = (i < N_DIM) ? (x[b * N_DIM + i] - parms[b * N_DIM + i])
                            : (-cvec[i - N_DIM]);
  }
  __syncthreads();

  // ---- forward: L (unit lower) ----
  for (int kb = 0; kb < NBLK; ++kb) {
    const int base = kb * NB;
    // stage diagonal block
    for (int idx = tid; idx < NB * NB; idx += 256) {
      int i = idx >> 6, j = idx & 63;
      sDg[i][j] = K[(base + i) * LDK + (base + j)];
    }
    __syncthreads();
    if (tid < 32) {
      const int cc = tid;
      for (int i = 1; i < NB; ++i) {
        float s = sX[base + i][cc];
        for (int j = 0; j < i; ++j) s -= sDg[i][j] * sX[base + j][cc];
        sX[base + i][cc] = s;
      }
    }
    __syncthreads();
    const int nt = NBLK - 1 - kb;
    for (int t = wave; t < nt * 8; t += 8) {
      const int blk = t >> 3;
      const int tt  = t & 7;
      const int ib  = (kb + 1 + blk) * NB;
      const int tr  = (tt >> 1) * 16;
      const int tc  = (tt & 1) * 16;
      v8f acc = tile_gemm(K + (ib + tr) * LDK + base, LDK,
                          &sX[base][tc], 33, m16, khi);
      #pragma unroll
      for (int v = 0; v < 8; ++v)
        sX[ib + tr + hi * 8 + v][tc + m16] -= acc[v];
    }
    __syncthreads();
  }

  // ---- backward: U ----
  for (int kb = NBLK - 1; kb >= 0; --kb) {
    const int base = kb * NB;
    for (int idx = tid; idx < NB * NB; idx += 256) {
      int i = idx >> 6, j = idx & 63;
      sDg[i][j] = K[(base + i) * LDK + (base + j)];
    }
    __syncthreads();
    if (tid < 32) {
      const int cc = tid;
      for (int i = NB - 1; i >= 0; --i) {
        float s = sX[base + i][cc];
        for (int j = i + 1; j < NB; ++j) s -= sDg[i][j] * sX[base + j][cc];
        sX[base + i][cc] = s / sDg[i][i];
      }
    }
    __syncthreads();
    for (int t = wave; t < kb * 8; t += 8) {
      const int blk = t >> 3;
      const int tt  = t & 7;
      const int ib  = blk * NB;
      const int tr  = (tt >> 1) * 16;
      const int tc  = (tt & 1) * 16;
      v8f acc = tile_gemm(K + (ib + tr) * LDK + base, LDK,
                          &sX[base][tc], 33, m16, khi);
      #pragma unroll
      for (int v = 0; v < 8; ++v)
        sX[ib + tr + hi * 8 + v][tc + m16] -= acc[v];
    }
    __syncthreads();
  }

  // primal part, transposed, to out (B x N row-major)
  for (int idx = tid; idx < N_DIM * 32; idx += 256) {
    const int i  = idx >> 5;
    const int cc = idx & 31;
    out[(col0 + cc) * N_DIM + i] = sX[i][cc];
  }
}

// ---------------------------------------------------------------------------
extern "C" void kernel_launch(void* const* d_in, const int* in_sizes, int n_in,
                              void* d_out, int out_size, void* d_ws, size_t ws_size,
                              hipStream_t stream) {
  const float* x     = (const float*)d_in[0];   // (B, N)
  const float* parms = (const float*)d_in[1];   // (B, PD) with PD == N
  const float* M     = (const float*)d_in[2];   // (N, N)
  const float* A0    = (const float*)d_in[3];   // (E, N)
  // d_in[4] (B0) is dead: F(0,0) == c
  const float* cvec  = (const float*)d_in[5];   // (E,)
  float* K   = (float*)d_ws;                    // 576*576 fp32 scratch
  float* out = (float*)d_out;                   // (B, N)

  build_kkt_kernel<<<(KDIM * KDIM + 255) / 256, 256, 0, stream>>>(M, A0, K);
  lu_factor_kernel<<<1, 1024, 0, stream>>>(K);
  kkt_solve_kernel<<<BATCH / 32, 256, 0, stream>>>(K, x, parms, cvec, out);
}